// InputLayer_14989435863704
// MI455X (gfx1250) — compile-verified
//
#include <hip/hip_runtime.h>
#include <cstdint>

// ---- problem constants (match reference) -----------------------------------
constexpr int Kc = 64;    // nodes per group
constexpr int Cc = 256;   // categories per node
constexpr int Bc = 512;   // batch

typedef __attribute__((ext_vector_type(4))) float        v4f;
typedef __attribute__((ext_vector_type(4))) int          v4i;
typedef __attribute__((ext_vector_type(4))) unsigned int v4u;
typedef __attribute__((ext_vector_type(8))) int          v8i;

#ifndef __has_builtin
#define __has_builtin(x) 0
#endif

#if __has_builtin(__builtin_amdgcn_tensor_load_to_lds) && \
    __has_builtin(__builtin_amdgcn_s_wait_tensorcnt)
#define USE_TDM 1
#else
#define USE_TDM 0
#endif

__device__ __forceinline__ void wait_async0() {
#if __has_builtin(__builtin_amdgcn_s_wait_asynccnt)
  __builtin_amdgcn_s_wait_asynccnt(0);
#else
  asm volatile("s_wait_asynccnt 0" ::: "memory");
#endif
}

// 16-byte global -> LDS copy on the gfx1250 async DMA path (ASYNCcnt).
__device__ __forceinline__ void async_ld_b128(const void* gptr, void* lptr) {
#if __has_builtin(__builtin_amdgcn_global_load_async_to_lds_b128)
  __builtin_amdgcn_global_load_async_to_lds_b128(
      (v4i*)gptr, (v4i*)lptr, /*offset=*/0, /*cpol=*/0);
#else
  *(v4i*)lptr = *(const v4i*)gptr;
#endif
}

#if USE_TDM
// One-shot TDM: DMA a 2-D tile (rows x cols elements of 4 B, row stride in
// elements) from global into LDS. D# packing per CDNA5 ISA §8.3/§8.4.
// This toolchain's builtin takes 6 operands: g0(v4u), g1(v8i), g2(v4i),
// g3(v4i), g4(v8i, zeroed for <=2-D), cpol.
__device__ __forceinline__ void tdm_load_2d_f32(const float* gsrc, void* lds,
                                                int rows, int cols, int row_stride) {
  const unsigned long long ga = (unsigned long long)(uintptr_t)gsrc;
  const unsigned lds_off = (unsigned)(uintptr_t)lds;  // low 32 bits == LDS byte offset

  v4u g0;
  g0[0] = 1u;                                        // count=1 (valid), flags 0
  g0[1] = lds_off;                                   // lds_addr
  g0[2] = (unsigned)(ga & 0xffffffffu);              // global_addr[31:0]
  g0[3] = (unsigned)((ga >> 32) & 0x01ffffffu)       // global_addr[56:32]
          | (2u << 30);                              // type = 2 ("image")

  v8i g1;
  g1[0] = 2 << 16;                                   // data_size=2 (4 B); wg_mask=0
  g1[1] = (cols & 0xffff) << 16;                     // tensor_dim0[15:0]
  g1[2] = ((cols >> 16) & 0xffff)                    // tensor_dim0[31:16]
          | ((rows & 0xffff) << 16);                 // tensor_dim1[15:0]
  g1[3] = ((rows >> 16) & 0xffff)                    // tensor_dim1[31:16]
          | ((cols & 0xffff) << 16);                 // tile_dim0 = cols
  g1[4] = rows & 0xffff;                             // tile_dim1 = rows; tile_dim2=0
  g1[5] = row_stride;                                // tensor_dim0_stride[31:0]
  g1[6] = 0;                                         // stride hi / dim1_stride lo
  g1[7] = 0;                                         // dim1_stride hi (unused, 2-D)

  v4i g2 = {0, 0, 0, 0};                             // 2-D tensor: groups 2/3 unused
  v4i g3 = {0, 0, 0, 0};
  v8i g4 = {0, 0, 0, 0, 0, 0, 0, 0};                 // extra group (clang-23 form): disabled
  __builtin_amdgcn_tensor_load_to_lds(g0, g1, g2, g3, g4, /*cpol=*/0);
}
#endif

__global__ __launch_bounds__(512, 1)
void cat_fw_mars_kernel(const int*   __restrict__ data,     // [V, B]
                        const float* __restrict__ params,   // flat
                        const int*   __restrict__ vids,     // [G]
                        const int*   __restrict__ s_pids,   // [G]
                        const int*   __restrict__ inc_pids, // [G]
                        float*       __restrict__ out)      // [G*K, B]
{
  __shared__ float p_lds[Kc * Cc];   // 64 KB: this group's param slice
  __shared__ int   d_lds[Bc];        //  2 KB: this group's observed categories

  const int g   = blockIdx.x;
  const int tid = threadIdx.x;       // 0..511

  const int vid = vids[g];
  const int sp  = s_pids[g];
  const int inc = inc_pids[g];

#if USE_TDM
  // ---- wave 0 issues one TDM descriptor for the whole 64 KB param slice ----
  if (tid < 32) {
    tdm_load_2d_f32(params + sp, p_lds, Kc, Cc, inc);
  }
#else
  // ---- fallback: lane-level async staging, fully coalesced b128 ------------
  #pragma unroll
  for (int i = 0; i < (Kc * Cc / 4) / 512; ++i) {
    const int flat = (tid + i * 512) << 2;
    const int k    = flat >> 8;
    const int c    = flat & (Cc - 1);
    async_ld_b128(params + sp + k * inc + c, p_lds + flat);
  }
#endif

  // ---- stage category row: 512 ints = 128 x b128 (first 4 waves) ----------
  if (tid < Bc / 4) {
    async_ld_b128(data + vid * Bc + tid * 4, d_lds + tid * 4);
  }

#if USE_TDM
  if (tid < 32) {
    __builtin_amdgcn_s_wait_tensorcnt(0);   // issuing wave drains TENSORcnt
  }
#endif
  wait_async0();       // each wave drains its own ASYNCcnt ...
  __syncthreads();     // ... barrier publishes all LDS writes to the block

  // ---- gather + log + coalesced NT stores ----------------------------------
  const int bq = tid & 127;          // float4 column index over B
  const int kb = tid >> 7;           // 0..3 : k phase
  const v4i c4 = ((const v4i*)d_lds)[bq];

  v4f* outv = (v4f*)out;
  const int gbase = g * (Kc * (Bc / 4));
  #pragma unroll
  for (int it = 0; it < Kc / 4; ++it) {
    const int    k   = it * 4 + kb;
    const float* row = p_lds + k * Cc;         // bank = category mod 64
    v4f r;
    r.x = __logf(row[c4.x]);
    r.y = __logf(row[c4.y]);
    r.z = __logf(row[c4.z]);
    r.w = __logf(row[c4.w]);
    // Output is write-once 256 MB: non-temporal so it doesn't thrash L2.
    __builtin_nontemporal_store(r, outv + (gbase + k * (Bc / 4) + bq));
  }
}

extern "C" void kernel_launch(void* const* d_in, const int* in_sizes, int n_in,
                              void* d_out, int out_size, void* d_ws, size_t ws_size,
                              hipStream_t stream) {
  const int*   data     = (const int*)  d_in[0];
  const float* params   = (const float*)d_in[1];
  const int*   vids     = (const int*)  d_in[2];
  const int*   s_pids   = (const int*)  d_in[3];
  const int*   inc_pids = (const int*)  d_in[4];
  float*       out      = (float*)d_out;

  const int Gn = in_sizes[2];   // number of groups = len(vids)

  cat_fw_mars_kernel<<<dim3((unsigned)Gn), dim3(512), 0, stream>>>(
      data, params, vids, s_pids, inc_pids, out);

  (void)n_in; (void)out_size; (void)d_ws; (void)ws_size;
}